// GCNGraph_73332271612104
// MI455X (gfx1250) — compile-verified
//
#include <hip/hip_runtime.h>

typedef __attribute__((ext_vector_type(2))) float v2f;
typedef __attribute__((ext_vector_type(8))) float v8f;

#define HID 64

// ---------------- degree / normalization ----------------
__global__ void k_deg_init(float* deg, int n) {
    int i = blockIdx.x * blockDim.x + threadIdx.x;
    if (i < n) deg[i] = 1.0f;   // self-loop contributes 1
}

__global__ void k_deg_edges(const int* __restrict__ dst, float* deg, int e) {
    int i = blockIdx.x * blockDim.x + threadIdx.x;
    if (i < e) atomicAdd(&deg[dst[i]], 1.0f);
}

__global__ void k_dinv(float* deg, int n) {
    int i = blockIdx.x * blockDim.x + threadIdx.x;
    if (i < n) {
        float d = deg[i];
        deg[i] = rsqrtf(d < 1.0f ? 1.0f : d);   // in-place: deg -> dinv
    }
}

// ---------------- fp32 GEMM via V_WMMA_F32_16X16X4_F32 ----------------
// Y[n,64] = X[n,64] @ W[64,64].  One wave per 16-row stripe; 4 N-tiles; 16 k-steps.
__global__ void k_gemm64_wmma(const float* __restrict__ X,
                              const float* __restrict__ W,
                              float* __restrict__ Y, int n) {
    int wave = (int)((blockIdx.x * (unsigned)blockDim.x + threadIdx.x) >> 5);
    int lane = threadIdx.x & 31;
    int ntiles = (n + 15) >> 4;
    if (wave >= ntiles) return;            // wave-uniform: EXEC stays all-ones

    int row0 = wave << 4;
    int m  = lane & 15;                    // matrix row (A) / column (B,C)
    int kh = lane >> 4;                    // K half-select (0: K0..1, 1: K2..3)
    int r  = row0 + m; if (r > n - 1) r = n - 1;   // clamp (no divergence)

    v8f acc0 = {}, acc1 = {}, acc2 = {}, acc3 = {};

    for (int k0 = 0; k0 < HID; k0 += 4) {
        int ka = k0 + 2 * kh;
        // A tile: lane holds X[row0+m][ka], X[row0+m][ka+1]
        const float2 axy = *reinterpret_cast<const float2*>(X + (size_t)r * HID + ka);
        v2f a; a.x = axy.x; a.y = axy.y;
        // B tiles: lane holds W[ka][nt*16+m], W[ka+1][nt*16+m]
        const float* w0 = W + (size_t)ka * HID + m;
        const float* w1 = W + (size_t)(ka + 1) * HID + m;
        v2f b0; b0.x = w0[0];  b0.y = w1[0];
        v2f b1; b1.x = w0[16]; b1.y = w1[16];
        v2f b2; b2.x = w0[32]; b2.y = w1[32];
        v2f b3; b3.x = w0[48]; b3.y = w1[48];

        acc0 = __builtin_amdgcn_wmma_f32_16x16x4_f32(false, a, false, b0, (short)0, acc0, false, false);
        acc1 = __builtin_amdgcn_wmma_f32_16x16x4_f32(false, a, false, b1, (short)0, acc1, false, false);
        acc2 = __builtin_amdgcn_wmma_f32_16x16x4_f32(false, a, false, b2, (short)0, acc2, false, false);
        acc3 = __builtin_amdgcn_wmma_f32_16x16x4_f32(false, a, false, b3, (short)0, acc3, false, false);
    }

    // C/D layout: VGPR rr -> M = rr + 8*kh, N = m (per 16-lane half)
    for (int rr = 0; rr < 8; ++rr) {
        int row = row0 + rr + 8 * kh;
        if (row < n) {
            float* yp = Y + (size_t)row * HID + m;
            yp[0]  = acc0[rr];
            yp[16] = acc1[rr];
            yp[32] = acc2[rr];
            yp[48] = acc3[rr];
        }
    }
}

// ---------------- bias init / scatter-add / relu ----------------
__global__ void k_init_bias(float* __restrict__ O, const float* __restrict__ b, int n) {
    long i = blockIdx.x * (long)blockDim.x + threadIdx.x;
    if (i < (long)n * HID) O[i] = b[i & (HID - 1)];
}

// Edge scatter: 16 lanes per edge, float4 gather of T[src], atomic add into O[dst].
// Edges [0,e) come from edge_index; [e, e+n) are self-loops.
__global__ void k_scatter(const int* __restrict__ src, const int* __restrict__ dst,
                          const float* __restrict__ dinv,
                          const float* __restrict__ T, float* __restrict__ O,
                          int e, int n) {
    long idx = blockIdx.x * (long)blockDim.x + threadIdx.x;
    int eid = (int)(idx >> 4);
    int sub = (int)(idx & 15);
    if (eid >= e + n) return;
    int s, d;
    if (eid < e) { s = src[eid]; d = dst[eid]; }
    else         { s = d = eid - e; }
    float nm = dinv[s] * dinv[d];
    float4 v = *reinterpret_cast<const float4*>(T + (size_t)s * HID + sub * 4);
    float* op = O + (size_t)d * HID + sub * 4;
    atomicAdd(op + 0, v.x * nm);
    atomicAdd(op + 1, v.y * nm);
    atomicAdd(op + 2, v.z * nm);
    atomicAdd(op + 3, v.w * nm);
}

__global__ void k_relu(float* __restrict__ O, long total) {
    long i = blockIdx.x * (long)blockDim.x + threadIdx.x;
    if (i < total) O[i] = fmaxf(O[i], 0.0f);
}

// ---------------- mean pool + classifier head ----------------
__global__ void k_pool_zero(float* sums, float* cnt, int gtot) {
    int i = blockIdx.x * blockDim.x + threadIdx.x;
    if (i < gtot * HID) sums[i] = 0.0f;
    if (i < gtot) cnt[i] = 0.0f;
}

__global__ void k_pool_accum(const int* __restrict__ batch, const float* __restrict__ H,
                             float* __restrict__ sums, float* __restrict__ cnt, int n) {
    long idx = blockIdx.x * (long)blockDim.x + threadIdx.x;
    int node = (int)(idx >> 4);
    int sub  = (int)(idx & 15);
    if (node >= n) return;
    int g = batch[node];
    float4 v = *reinterpret_cast<const float4*>(H + (size_t)node * HID + sub * 4);
    float* sp = sums + (size_t)g * HID + sub * 4;
    atomicAdd(sp + 0, v.x);
    atomicAdd(sp + 1, v.y);
    atomicAdd(sp + 2, v.z);
    atomicAdd(sp + 3, v.w);
    if (sub == 0) atomicAdd(&cnt[g], 1.0f);
}

__global__ void k_final(const float* __restrict__ sums, const float* __restrict__ cnt,
                        const float* __restrict__ Wlin, const float* __restrict__ blin,
                        float* __restrict__ out, int G, int C) {
    int i = blockIdx.x * blockDim.x + threadIdx.x;
    if (i >= G * C) return;
    int g = i / C, c = i % C;
    float cn = cnt[g]; if (cn < 1.0f) cn = 1.0f;
    float acc = 0.0f;
    for (int k = 0; k < HID; ++k) acc += sums[(size_t)g * HID + k] * Wlin[(size_t)k * C + c];
    out[i] = acc / cn + blin[c];
}

// ---------------- host-side orchestration ----------------
extern "C" void kernel_launch(void* const* d_in, const int* in_sizes, int n_in,
                              void* d_out, int out_size, void* d_ws, size_t ws_size,
                              hipStream_t stream) {
    const float* x    = (const float*)d_in[0];
    const int*   ei   = (const int*)  d_in[1];
    const int*   bat  = (const int*)  d_in[2];
    const float* W1   = (const float*)d_in[3];
    const float* b1   = (const float*)d_in[4];
    const float* W2   = (const float*)d_in[5];
    const float* b2   = (const float*)d_in[6];
    const float* Wlin = (const float*)d_in[7];
    const float* blin = (const float*)d_in[8];
    float* out = (float*)d_out;

    const int n = in_sizes[2];          // N_NODES (batch has one entry per node)
    const int e = in_sizes[1] / 2;      // edge_index is (2, E)
    const int C = in_sizes[8];          // NUM_CLASSES
    const int G = 64;                   // NUM_GRAPHS (fixed by reference)

    const int* src = ei;
    const int* dst = ei + e;

    float* ws   = (float*)d_ws;
    float* dinv = ws;                                   // n floats (deg -> dinv in place)
    size_t off  = ((size_t)n + 255) & ~(size_t)255;
    float* bufT = ws + off;                             // n*64
    float* bufH = bufT + (size_t)n * HID;               // n*64
    float* sums = bufH + (size_t)n * HID;               // G*64
    float* cnts = sums + (size_t)G * HID;               // G

    const int TB = 256;
    const long nElem = (long)n * HID;
    dim3 blkNodes((n + TB - 1) / TB);
    dim3 blkEdges((e + TB - 1) / TB);
    dim3 blkElem((unsigned)((nElem + TB - 1) / TB));
    int ntiles = (n + 15) >> 4;
    dim3 blkGemm((ntiles + 7) / 8);                     // 8 waves / 256-thread block
    long scatterItems = (long)(e + n) * 16;
    dim3 blkScat((unsigned)((scatterItems + TB - 1) / TB));
    dim3 blkPool((unsigned)(((long)n * 16 + TB - 1) / TB));

    // normalization: deg -> dinv
    k_deg_init <<<blkNodes, TB, 0, stream>>>(dinv, n);
    k_deg_edges<<<blkEdges, TB, 0, stream>>>(dst, dinv, e);
    k_dinv     <<<blkNodes, TB, 0, stream>>>(dinv, n);

    // layer 1
    k_gemm64_wmma<<<blkGemm, TB, 0, stream>>>(x, W1, bufT, n);
    k_init_bias  <<<blkElem, TB, 0, stream>>>(bufH, b1, n);
    k_scatter    <<<blkScat, TB, 0, stream>>>(src, dst, dinv, bufT, bufH, e, n);
    k_relu       <<<blkElem, TB, 0, stream>>>(bufH, nElem);

    // layer 2
    k_gemm64_wmma<<<blkGemm, TB, 0, stream>>>(bufH, W2, bufT, n);
    k_init_bias  <<<blkElem, TB, 0, stream>>>(bufH, b2, n);
    k_scatter    <<<blkScat, TB, 0, stream>>>(src, dst, dinv, bufT, bufH, e, n);
    k_relu       <<<blkElem, TB, 0, stream>>>(bufH, nElem);

    // mean pool + linear head
    k_pool_zero <<<(G * HID + TB - 1) / TB, TB, 0, stream>>>(sums, cnts, G);
    k_pool_accum<<<blkPool, TB, 0, stream>>>(bat, bufH, sums, cnts, n);
    k_final     <<<(G * C + TB - 1) / TB, TB, 0, stream>>>(sums, cnts, Wlin, blin, out, G, C);
}